// FlexibleLogisticModel_79302276153846
// MI455X (gfx1250) — compile-verified
//
#include <hip/hip_runtime.h>
#include <math.h>

// CDNA5 (gfx1250) wave32. WMMA f32 16x16x4: A = 16x4 f32 (2 VGPR/lane), B = 4x16 f32,
// C/D = 16x16 f32 (8 VGPR/lane).
typedef __attribute__((ext_vector_type(2))) float v2f;
typedef __attribute__((ext_vector_type(8))) float v8f;

#define N_TRIPLE 45760   // C(66,3): degree-4 triple prefixes (i<=j<=k)
#define N_PAIR   2080    // C(65,2): degree-3/2 pair prefixes (i<=j)
#define OFF1 1
#define OFF2 65
#define OFF3 2145
#define OFF4 47905
#define C4TOT 766480     // C(67,4)
#define C3TOT 45760      // C(66,3)
#define C2TOT 2080       // C(65,2)

__device__ __forceinline__ int C2i(int n) { return (n * (n - 1)) >> 1; }
__device__ __forceinline__ int C3i(int n) { return n * (n - 1) * (n - 2) / 6; }
__device__ __forceinline__ int C4i(int n) { return n * (n - 1) * (n - 2) * (n - 3) / 24; }

__global__ void poly_partial_kernel(const float* __restrict__ x,
                                    const float* __restrict__ w,
                                    const float* __restrict__ alphas,
                                    float* __restrict__ blockPartial) {
    __shared__ float xs[64];
    __shared__ float wsum[8];
    const int tid = threadIdx.x;
    if (tid < 64) xs[tid] = x[tid];
    __syncthreads();

    const int t = blockIdx.x * blockDim.x + tid;
    float p = 0.f;

    if (t < N_TRIPLE) {
        // ---- degree-4: unrank triple prefix (i<=j<=k), loop l = k..63 (contiguous w) ----
        int r = t, i = 0;
        while (r >= C2i(65 - i)) { r -= C2i(65 - i); ++i; }   // #triples with first elem i = C(65-i,2)
        int j = i;
        while (r >= 64 - j) { r -= 64 - j; ++j; }             // #triples (i,j,*) = 64-j
        const int k = j + r;
        const float q = xs[i] * xs[j] * xs[k];
        int idx = OFF4 + (C4TOT - C4i(67 - i)) + (C3i(66 - i) - C3i(66 - j))
                       + (C2i(65 - j) - C2i(65 - k));         // rank4(i,j,k,k)
        float s = 0.f;
        for (int l = k; l < 64; ++l) s += w[idx + (l - k)] * xs[l];
        p = alphas[4] * q * s;
    } else if (t < N_TRIPLE + N_PAIR) {
        // ---- degree-3 + degree-2: unrank pair (i<=j) ----
        const int pr = t - N_TRIPLE;
        int r = pr, i = 0;
        while (r >= 64 - i) { r -= 64 - i; ++i; }
        const int j = i + r;
        const float q = xs[i] * xs[j];
        const int idx2 = OFF2 + (C2TOT - C2i(65 - i)) + (j - i);
        p = alphas[2] * w[idx2] * q;
        int idx3 = OFF3 + (C3TOT - C3i(66 - i)) + (C2i(65 - i) - C2i(65 - j)); // rank3(i,j,j)
        float s = 0.f;
        for (int k = j; k < 64; ++k) s += w[idx3 + (k - j)] * xs[k];
        p += alphas[3] * q * s;
        if (pr == 0) {
            // ---- degrees 1 and 0 ----
            float s1 = 0.f;
            for (int d = 0; d < 64; ++d) s1 += w[OFF1 + d] * xs[d];
            p += alphas[1] * s1 + alphas[0] * w[0];
        }
    }

    // ---- wave32 reduction on the matrix pipe (full f32 precision) ----
    // A[m,0]=p_m (lanes 0-15), A[m,2]=p_{16+m} (lanes 16-31); B = all ones.
    // D[m,j] = p_m + p_{16+m}; then sum the 8 accumulator VGPRs in-lane and fold halves.
    v2f a; a.x = p;  a.y = 0.f;
    v2f b; b.x = 1.f; b.y = 1.f;
    v8f c = {};
    c = __builtin_amdgcn_wmma_f32_16x16x4_f32(false, a, false, b, (short)0, c, false, false);
    float s = c[0] + c[1] + c[2] + c[3] + c[4] + c[5] + c[6] + c[7];
    s += __shfl_xor(s, 16, 32);   // all 32 lanes now hold the wave total

    const int lane = tid & 31, wid = tid >> 5;
    if (lane == 0) wsum[wid] = s;
    __syncthreads();
    if (tid == 0) {
        float acc = 0.f;
#pragma unroll
        for (int i2 = 0; i2 < 8; ++i2) acc += wsum[i2];
        blockPartial[blockIdx.x] = acc;   // deterministic: no float atomics
    }
}

__global__ void poly_finalize_kernel(const float* __restrict__ blockPartial,
                                     int nblk, float* __restrict__ out) {
    __shared__ float buf[256];
    float s = 0.f;
    for (int i = threadIdx.x; i < nblk; i += 256) s += blockPartial[i];
    buf[threadIdx.x] = s;
    __syncthreads();
    for (int off = 128; off > 0; off >>= 1) {
        if ((int)threadIdx.x < off) buf[threadIdx.x] += buf[threadIdx.x + off];
        __syncthreads();
    }
    if (threadIdx.x == 0) out[0] = 1.f / (1.f + expf(-buf[0]));
}

extern "C" void kernel_launch(void* const* d_in, const int* in_sizes, int n_in,
                              void* d_out, int out_size, void* d_ws, size_t ws_size,
                              hipStream_t stream) {
    const float* x      = (const float*)d_in[0];  // [64]
    const float* w      = (const float*)d_in[1];  // [814385]
    const float* alphas = (const float*)d_in[2];  // [5]
    // d_in[3] (E, 208 MB) and d_in[4] (ord_ids) intentionally unused:
    // the exponent structure is computed analytically (combinatorial unranking).
    float* out     = (float*)d_out;
    float* partial = (float*)d_ws;   // nblk floats of scratch

    const int total = N_TRIPLE + N_PAIR;      // 47840 threads
    const int block = 256;                    // 8 wave32 per block
    const int nblk  = (total + block - 1) / block;  // 187

    poly_partial_kernel<<<nblk, block, 0, stream>>>(x, w, alphas, partial);
    poly_finalize_kernel<<<1, 256, 0, stream>>>(partial, nblk, out);
}